// GAT_35459249995965
// MI455X (gfx1250) — compile-verified
//
#include <hip/hip_runtime.h>
#include <math.h>

#define NN 100000
#define EE 1600000
#define ET (EE + NN)
#define HH 4
#define CC 32
#define HC 128
#define OUTC 40
#define OUTCP 48                        // OUTC padded to multiple of 16
#define NEG_SLOPE 0.2f
#define BN_SCALE 0.9999950000374997f    // 1/sqrt(1+1e-5)

typedef __attribute__((ext_vector_type(2))) float v2f;
typedef __attribute__((ext_vector_type(8))) float v8f;

__device__ __forceinline__ unsigned f2ord(float f) {
  unsigned u = __float_as_uint(f);
  return (u & 0x80000000u) ? ~u : (u | 0x80000000u);
}
__device__ __forceinline__ float ord2f(unsigned u) {
  return (u & 0x80000000u) ? __uint_as_float(u & 0x7fffffffu) : __uint_as_float(~u);
}
__device__ __forceinline__ float lrelu(float x) { return x > 0.f ? x : NEG_SLOPE * x; }

// ---------------------------------------------------------------- fill
__global__ void fill_u32(unsigned* __restrict__ p, unsigned v, int n) {
  int i = blockIdx.x * blockDim.x + threadIdx.x;
  int stride = gridDim.x * blockDim.x;
  for (; i < n; i += stride) p[i] = v;
}

// ---------------------------------------------------------------- weight transpose + pad
// W[K,Ncol] -> WT[NcolPad,K], zero-filled for col >= Ncol.
__global__ void transpose_pad(const float* __restrict__ W, float* __restrict__ WT,
                              int K, int Ncol, int NcolPad) {
  int i = blockIdx.x * blockDim.x + threadIdx.x;
  if (i >= K * NcolPad) return;
  int col = i / K;
  int k   = i - col * K;
  WT[i] = (col < Ncol) ? W[(size_t)k * Ncol + col] : 0.f;
}

// ---------------------------------------------------------------- fp32 WMMA GEMM
// C[M,NcolC] = A[M,K] @ B[K,*] with B given transposed/padded as BT[NcolPad,K].
// One wave per 16x16 output tile; inner loop: 2x global_load_b64 + 1x v_wmma.
__global__ void wmma_gemm(const float* __restrict__ A, const float* __restrict__ BT,
                          float* __restrict__ C, int M, int K, int NcolC, int ntN) {
  int wave = (blockIdx.x * blockDim.x + threadIdx.x) >> 5;
  int lane = threadIdx.x & 31;
  int mt = wave / ntN;
  int nt = wave - mt * ntN;
  if (mt * 16 >= M) return;
  int half = lane >> 4;        // 0: K pair {0,1}; 1: K pair {2,3}
  int l16  = lane & 15;
  const float* ap = A  + (size_t)(mt * 16 + l16) * K + half * 2;
  const float* bp = BT + (size_t)(nt * 16 + l16) * K + half * 2;
  v8f acc = {};
#pragma unroll 4
  for (int k0 = 0; k0 < K; k0 += 4) {
    v2f a = *reinterpret_cast<const v2f*>(ap + k0);
    v2f b = *reinterpret_cast<const v2f*>(bp + k0);
    acc = __builtin_amdgcn_wmma_f32_16x16x4_f32(false, a, false, b,
                                                (short)0, acc, false, false);
  }
  int bcol = nt * 16 + l16;
  if (bcol < NcolC) {
#pragma unroll
    for (int j = 0; j < 8; ++j) {
      int row = mt * 16 + half * 8 + j;
      C[(size_t)row * NcolC + bcol] = acc[j];
    }
  }
}

// ---------------------------------------------------------------- attention coefficients, H=4, C=32
__global__ void attn_coef(const float* __restrict__ h, const float* __restrict__ att_s,
                          const float* __restrict__ att_d,
                          float* __restrict__ asrc, float* __restrict__ adst) {
  int wave = (blockIdx.x * blockDim.x + threadIdx.x) >> 5;
  int lane = threadIdx.x & 31;
  if (wave >= NN) return;
  float ps[HH], pd[HH];
#pragma unroll
  for (int j = 0; j < HH; ++j) {
    float v = h[(size_t)wave * HC + j * CC + lane];
    ps[j] = v * att_s[j * CC + lane];
    pd[j] = v * att_d[j * CC + lane];
  }
#pragma unroll
  for (int off = 16; off > 0; off >>= 1) {
#pragma unroll
    for (int j = 0; j < HH; ++j) {
      ps[j] += __shfl_xor(ps[j], off, 32);
      pd[j] += __shfl_xor(pd[j], off, 32);
    }
  }
  if (lane == 0) {
#pragma unroll
    for (int j = 0; j < HH; ++j) {
      asrc[wave * HH + j] = ps[j];
      adst[wave * HH + j] = pd[j];
    }
  }
}

// attention coefficients, H=1, C=40 (layer 2)
__global__ void attn_coef_l2(const float* __restrict__ h, const float* __restrict__ att_s,
                             const float* __restrict__ att_d,
                             float* __restrict__ asrc, float* __restrict__ adst) {
  int wave = (blockIdx.x * blockDim.x + threadIdx.x) >> 5;
  int lane = threadIdx.x & 31;
  if (wave >= NN) return;
  float v0 = h[(size_t)wave * OUTC + lane];
  float ps = v0 * att_s[lane];
  float pd = v0 * att_d[lane];
  if (lane < OUTC - 32) {
    float v1 = h[(size_t)wave * OUTC + 32 + lane];
    ps += v1 * att_s[32 + lane];
    pd += v1 * att_d[32 + lane];
  }
#pragma unroll
  for (int off = 16; off > 0; off >>= 1) {
    ps += __shfl_xor(ps, off, 32);
    pd += __shfl_xor(pd, off, 32);
  }
  if (lane == 0) { asrc[wave] = ps; adst[wave] = pd; }
}

// ---------------------------------------------------------------- edge passes, H=4
__global__ void edge_max(const int* __restrict__ src, const int* __restrict__ dst,
                         const float* __restrict__ asrc, const float* __restrict__ adst,
                         unsigned* __restrict__ m) {
  int e = blockIdx.x * blockDim.x + threadIdx.x;
  if (e >= ET) return;
  int s, d;
  if (e < EE) { s = src[e]; d = dst[e]; } else { s = d = e - EE; }
  const float4 A = reinterpret_cast<const float4*>(asrc)[s];
  const float4 D = reinterpret_cast<const float4*>(adst)[d];
  atomicMax(&m[d * HH + 0], f2ord(lrelu(A.x + D.x)));
  atomicMax(&m[d * HH + 1], f2ord(lrelu(A.y + D.y)));
  atomicMax(&m[d * HH + 2], f2ord(lrelu(A.z + D.z)));
  atomicMax(&m[d * HH + 3], f2ord(lrelu(A.w + D.w)));
}

__global__ void edge_sum(const int* __restrict__ src, const int* __restrict__ dst,
                         const float* __restrict__ asrc, const float* __restrict__ adst,
                         const unsigned* __restrict__ m, float* __restrict__ sb) {
  int e = blockIdx.x * blockDim.x + threadIdx.x;
  if (e >= ET) return;
  int s, d;
  if (e < EE) { s = src[e]; d = dst[e]; } else { s = d = e - EE; }
  const float4 A = reinterpret_cast<const float4*>(asrc)[s];
  const float4 D = reinterpret_cast<const float4*>(adst)[d];
  atomicAdd(&sb[d * HH + 0], expf(lrelu(A.x + D.x) - ord2f(m[d * HH + 0])));
  atomicAdd(&sb[d * HH + 1], expf(lrelu(A.y + D.y) - ord2f(m[d * HH + 1])));
  atomicAdd(&sb[d * HH + 2], expf(lrelu(A.z + D.z) - ord2f(m[d * HH + 2])));
  atomicAdd(&sb[d * HH + 3], expf(lrelu(A.w + D.w) - ord2f(m[d * HH + 3])));
}

__global__ void edge_aggr(const int* __restrict__ src, const int* __restrict__ dst,
                          const float* __restrict__ asrc, const float* __restrict__ adst,
                          const unsigned* __restrict__ m, const float* __restrict__ sb,
                          const float* __restrict__ h, float* __restrict__ acc) {
  int wave = (blockIdx.x * blockDim.x + threadIdx.x) >> 5;
  int lane = threadIdx.x & 31;
  if (wave >= ET) return;
  int s, d;
  if (wave < EE) { s = src[wave]; d = dst[wave]; } else { s = d = wave - EE; }
  const float4 A = reinterpret_cast<const float4*>(asrc)[s];
  const float4 D = reinterpret_cast<const float4*>(adst)[d];
  float al[HH];
  {
    float ax[HH] = {A.x + D.x, A.y + D.y, A.z + D.z, A.w + D.w};
#pragma unroll
    for (int j = 0; j < HH; ++j) {
      float mm = ord2f(m[d * HH + j]);
      float ss = sb[d * HH + j];
      al[j] = expf(lrelu(ax[j]) - mm) / (ss + 1e-16f);
    }
  }
#pragma unroll
  for (int j = 0; j < HH; ++j) {
    int col = j * CC + lane;
    atomicAdd(&acc[(size_t)d * HC + col], h[(size_t)s * HC + col] * al[j]);
  }
}

// ---------------------------------------------------------------- edge passes, H=1 (layer 2)
__global__ void edge_max_l2(const int* __restrict__ src, const int* __restrict__ dst,
                            const float* __restrict__ asrc, const float* __restrict__ adst,
                            unsigned* __restrict__ m) {
  int e = blockIdx.x * blockDim.x + threadIdx.x;
  if (e >= ET) return;
  int s, d;
  if (e < EE) { s = src[e]; d = dst[e]; } else { s = d = e - EE; }
  atomicMax(&m[d], f2ord(lrelu(asrc[s] + adst[d])));
}

__global__ void edge_sum_l2(const int* __restrict__ src, const int* __restrict__ dst,
                            const float* __restrict__ asrc, const float* __restrict__ adst,
                            const unsigned* __restrict__ m, float* __restrict__ sb) {
  int e = blockIdx.x * blockDim.x + threadIdx.x;
  if (e >= ET) return;
  int s, d;
  if (e < EE) { s = src[e]; d = dst[e]; } else { s = d = e - EE; }
  atomicAdd(&sb[d], expf(lrelu(asrc[s] + adst[d]) - ord2f(m[d])));
}

__global__ void edge_aggr_l2(const int* __restrict__ src, const int* __restrict__ dst,
                             const float* __restrict__ asrc, const float* __restrict__ adst,
                             const unsigned* __restrict__ m, const float* __restrict__ sb,
                             const float* __restrict__ h, float* __restrict__ acc) {
  int wave = (blockIdx.x * blockDim.x + threadIdx.x) >> 5;
  int lane = threadIdx.x & 31;
  if (wave >= ET) return;
  int s, d;
  if (wave < EE) { s = src[wave]; d = dst[wave]; } else { s = d = wave - EE; }
  float alpha = expf(lrelu(asrc[s] + adst[d]) - ord2f(m[d])) / (sb[d] + 1e-16f);
  atomicAdd(&acc[(size_t)d * OUTC + lane], h[(size_t)s * OUTC + lane] * alpha);
  if (lane < OUTC - 32)
    atomicAdd(&acc[(size_t)d * OUTC + 32 + lane], h[(size_t)s * OUTC + 32 + lane] * alpha);
}

// ---------------------------------------------------------------- bias + eval-BN + ReLU (in place)
__global__ void bn_relu(float* __restrict__ a, const float* __restrict__ bias,
                        const float* __restrict__ g, const float* __restrict__ be) {
  size_t i = (size_t)blockIdx.x * blockDim.x + threadIdx.x;
  if (i >= (size_t)NN * HC) return;
  int c = (int)(i & (HC - 1));
  float v = (a[i] + bias[c]) * BN_SCALE;
  v = g[c] * v + be[c];
  a[i] = v > 0.f ? v : 0.f;
}

// ---------------------------------------------------------------- +bias +blast, log_softmax over 40
__global__ void final_logsoftmax(const float* __restrict__ acc2, const float* __restrict__ b2,
                                 const float* __restrict__ blast, float* __restrict__ out) {
  int wave = (blockIdx.x * blockDim.x + threadIdx.x) >> 5;
  int lane = threadIdx.x & 31;
  if (wave >= NN) return;
  float v0 = acc2[(size_t)wave * OUTC + lane] + b2[lane] + blast[lane];
  float v1 = -INFINITY;
  if (lane < OUTC - 32)
    v1 = acc2[(size_t)wave * OUTC + 32 + lane] + b2[32 + lane] + blast[32 + lane];
  float mx = v0 > v1 ? v0 : v1;
#pragma unroll
  for (int off = 16; off > 0; off >>= 1) {
    float o = __shfl_xor(mx, off, 32);
    mx = mx > o ? mx : o;
  }
  float se = expf(v0 - mx) + (lane < OUTC - 32 ? expf(v1 - mx) : 0.f);
#pragma unroll
  for (int off = 16; off > 0; off >>= 1) se += __shfl_xor(se, off, 32);
  float lse = mx + logf(se);
  out[(size_t)wave * OUTC + lane] = v0 - lse;
  if (lane < OUTC - 32) out[(size_t)wave * OUTC + 32 + lane] = v1 - lse;
}

// ================================================================ host
extern "C" void kernel_launch(void* const* d_in, const int* in_sizes, int n_in,
                              void* d_out, int out_size, void* d_ws, size_t ws_size,
                              hipStream_t stream) {
  const float* x    = (const float*)d_in[0];
  const int*   ei   = (const int*)d_in[1];
  const float* W0   = (const float*)d_in[2];
  const float* as0  = (const float*)d_in[3];
  const float* ad0  = (const float*)d_in[4];
  const float* b0   = (const float*)d_in[5];
  const float* g0   = (const float*)d_in[6];
  const float* be0  = (const float*)d_in[7];
  const float* W1   = (const float*)d_in[8];
  const float* as1  = (const float*)d_in[9];
  const float* ad1  = (const float*)d_in[10];
  const float* b1   = (const float*)d_in[11];
  const float* g1   = (const float*)d_in[12];
  const float* be1  = (const float*)d_in[13];
  const float* W2   = (const float*)d_in[14];
  const float* as2  = (const float*)d_in[15];
  const float* ad2  = (const float*)d_in[16];
  const float* b2   = (const float*)d_in[17];
  const float* blast= (const float*)d_in[18];
  float* out = (float*)d_out;

  float* hbuf = (float*)d_ws;                           // N*128
  float* acc  = hbuf + (size_t)NN * HC;                 // N*128
  float* asrc = acc  + (size_t)NN * HC;                 // N*4
  float* adst = asrc + (size_t)NN * HH;                 // N*4
  unsigned* mbuf = (unsigned*)(adst + (size_t)NN * HH); // N*4
  float* sbuf = (float*)(mbuf + (size_t)NN * HH);       // N*4
  float* acc2 = sbuf + (size_t)NN * HH;                 // N*40
  float* wt0  = acc2 + (size_t)NN * OUTC;               // 128*128
  float* wt1  = wt0 + HC * HC;                          // 128*128
  float* wt2  = wt1 + HC * HC;                          // 48*128

  const int* srcp = ei;
  const int* dstp = ei + EE;
  const unsigned ORD_NINF = 0x007fffffu;                // f2ord(-inf)

  const int gemm_blocks  = (NN / 16) * 8 / 8;           // 6250 blocks x 8 waves
  const int gemm_blocks2 = ((NN / 16) * 3 + 7) / 8;     // layer-2 (3 col tiles)
  const int nodew_blocks = (NN * 32 + 255) / 256;       // 1 wave/node
  const int edge_blocks  = (ET + 255) / 256;            // 1 thread/edge
  const int edgew_blocks = ET / 8;                      // 1 wave/edge
  const int elem_blocks  = (NN * HC + 255) / 256;

  // ---- transpose/pad all weights once per call
  transpose_pad<<<(HC * HC + 255) / 256, 256, 0, stream>>>(W0, wt0, HC, HC, HC);
  transpose_pad<<<(HC * HC + 255) / 256, 256, 0, stream>>>(W1, wt1, HC, HC, HC);
  transpose_pad<<<(HC * OUTCP + 255) / 256, 256, 0, stream>>>(W2, wt2, HC, OUTC, OUTCP);

  // ---- layer 0
  wmma_gemm<<<gemm_blocks, 256, 0, stream>>>(x, wt0, hbuf, NN, HC, HC, 8);
  attn_coef<<<nodew_blocks, 256, 0, stream>>>(hbuf, as0, ad0, asrc, adst);
  fill_u32<<<1024, 256, 0, stream>>>(mbuf, ORD_NINF, NN * HH);
  fill_u32<<<1024, 256, 0, stream>>>((unsigned*)sbuf, 0u, NN * HH);
  fill_u32<<<4096, 256, 0, stream>>>((unsigned*)acc, 0u, NN * HC);
  edge_max <<<edge_blocks, 256, 0, stream>>>(srcp, dstp, asrc, adst, mbuf);
  edge_sum <<<edge_blocks, 256, 0, stream>>>(srcp, dstp, asrc, adst, mbuf, sbuf);
  edge_aggr<<<edgew_blocks, 256, 0, stream>>>(srcp, dstp, asrc, adst, mbuf, sbuf, hbuf, acc);
  bn_relu<<<elem_blocks, 256, 0, stream>>>(acc, b0, g0, be0);

  // ---- layer 1 (input = acc; after GEMM reads it, reuse as accumulator)
  wmma_gemm<<<gemm_blocks, 256, 0, stream>>>(acc, wt1, hbuf, NN, HC, HC, 8);
  attn_coef<<<nodew_blocks, 256, 0, stream>>>(hbuf, as1, ad1, asrc, adst);
  fill_u32<<<1024, 256, 0, stream>>>(mbuf, ORD_NINF, NN * HH);
  fill_u32<<<1024, 256, 0, stream>>>((unsigned*)sbuf, 0u, NN * HH);
  fill_u32<<<4096, 256, 0, stream>>>((unsigned*)acc, 0u, NN * HC);
  edge_max <<<edge_blocks, 256, 0, stream>>>(srcp, dstp, asrc, adst, mbuf);
  edge_sum <<<edge_blocks, 256, 0, stream>>>(srcp, dstp, asrc, adst, mbuf, sbuf);
  edge_aggr<<<edgew_blocks, 256, 0, stream>>>(srcp, dstp, asrc, adst, mbuf, sbuf, hbuf, acc);
  bn_relu<<<elem_blocks, 256, 0, stream>>>(acc, b1, g1, be1);

  // ---- layer 2 (heads=1, out=40)
  wmma_gemm<<<gemm_blocks2, 256, 0, stream>>>(acc, wt2, hbuf, NN, HC, OUTC, 3);
  attn_coef_l2<<<nodew_blocks, 256, 0, stream>>>(hbuf, as2, ad2, asrc, adst);
  fill_u32<<<512, 256, 0, stream>>>(mbuf, ORD_NINF, NN);
  fill_u32<<<512, 256, 0, stream>>>((unsigned*)sbuf, 0u, NN);
  fill_u32<<<2048, 256, 0, stream>>>((unsigned*)acc2, 0u, NN * OUTC);
  edge_max_l2 <<<edge_blocks, 256, 0, stream>>>(srcp, dstp, asrc, adst, mbuf);
  edge_sum_l2 <<<edge_blocks, 256, 0, stream>>>(srcp, dstp, asrc, adst, mbuf, sbuf);
  edge_aggr_l2<<<edgew_blocks, 256, 0, stream>>>(srcp, dstp, asrc, adst, mbuf, sbuf, hbuf, acc2);
  final_logsoftmax<<<nodew_blocks, 256, 0, stream>>>(acc2, b2, blast, out);
}